// RN_58497454572113
// MI455X (gfx1250) — compile-verified
//
#include <hip/hip_runtime.h>
#include <hip/hip_bf16.h>

// ---------------------------------------------------------------------------
// CLEVR Relation Network forward, fp32, gfx1250.
//
// Pipeline:
//   conv1..conv4 (+BN+ReLU, direct, VALU)  -> c4 [64,24,8,8]
//   assemble objects + coords (pad K 26->28) -> obj [4096,28]
//   LSTM (VALU, one block per batch)        -> qh  [64,128]
//   WMMA f32 16x16x4 GEMMs:
//     A = obj @ Wg[:, 0:26]^T   [4096,256]
//     B = obj @ Wg[:,26:52]^T   [4096,256]
//     Qc = qh @ Wg[:,52:180]^T + g0_b  [64,256]
//   pair reduce: xg[b,o] = sum_{i,j} relu(A[b,i,o]+B[b,j,o]+Qc[b,o])
//   f1,f2 (WMMA + bias + relu), f3 + log_softmax (wave32 shuffle reduce)
// ---------------------------------------------------------------------------

typedef float v2f __attribute__((ext_vector_type(2)));
typedef float v8f __attribute__((ext_vector_type(8)));

// ------------------------- conv + BN + ReLU (direct) -----------------------
__global__ void conv_bn_relu(const float* __restrict__ in, const float* __restrict__ w,
                             const float* __restrict__ cbias,
                             const float* __restrict__ bng, const float* __restrict__ bnb,
                             const float* __restrict__ bnm, const float* __restrict__ bnv,
                             float* __restrict__ out,
                             int IC, int OC, int IH, int IW, int OH, int OW, int NB) {
    int idx = blockIdx.x * blockDim.x + threadIdx.x;
    int total = NB * OC * OH * OW;
    if (idx >= total) return;
    int ox = idx % OW; int t = idx / OW;
    int oy = t % OH;   t /= OH;
    int oc = t % OC;   int nb = t / OC;

    const float* wbase = w + oc * IC * 9;
    const float* ibase = in + (size_t)nb * IC * IH * IW;
    int iy0 = oy * 2 - 1, ix0 = ox * 2 - 1;
    float acc = 0.f;
    for (int ic = 0; ic < IC; ++ic) {
        const float* ip = ibase + (size_t)ic * IH * IW;
        const float* wp = wbase + ic * 9;
#pragma unroll
        for (int ky = 0; ky < 3; ++ky) {
            int iy = iy0 + ky;
            if (iy < 0 || iy >= IH) continue;
#pragma unroll
            for (int kx = 0; kx < 3; ++kx) {
                int ix = ix0 + kx;
                if (ix < 0 || ix >= IW) continue;
                acc = fmaf(ip[iy * IW + ix], wp[ky * 3 + kx], acc);
            }
        }
    }
    float scale = bng[oc] * rsqrtf(bnv[oc] + 1e-5f);
    float val = (acc + cbias[oc]) * scale + (bnb[oc] - bnm[oc] * scale);
    out[idx] = fmaxf(val, 0.f);
}

// ----------------- assemble [4096,28] object features + coords -------------
__global__ void assemble_obj(const float* __restrict__ c4, float* __restrict__ obj) {
    int idx = blockIdx.x * blockDim.x + threadIdx.x;
    if (idx >= 4096 * 28) return;
    int c = idx % 28;
    int s = (idx / 28) % 64;   // spatial 0..63 (y*8+x)
    int b = idx / (28 * 64);
    float v;
    if (c < 24)       v = c4[(b * 24 + c) * 64 + s];
    else if (c == 24) v = -4.f + (8.f / 7.f) * (float)(s % 8);  // x coord
    else if (c == 25) v = -4.f + (8.f / 7.f) * (float)(s / 8);  // y coord
    else              v = 0.f;                                   // K pad
    obj[idx] = v;
}

// ---------------- pad g0_w column slices to K=28 (zeros in 26,27) ----------
__global__ void pad_gw(const float* __restrict__ g0_w,
                       float* __restrict__ wga, float* __restrict__ wgb) {
    int idx = blockIdx.x * blockDim.x + threadIdx.x;
    if (idx >= 256 * 28) return;
    int c = idx % 28, o = idx / 28;
    wga[idx] = (c < 26) ? g0_w[o * 180 + c]      : 0.f;
    wgb[idx] = (c < 26) ? g0_w[o * 180 + 26 + c] : 0.f;
}

// ------------------------------- LSTM (VALU) -------------------------------
__global__ void lstm_kernel(const int* __restrict__ qidx, const float* __restrict__ emb,
                            const float* __restrict__ w_ih, const float* __restrict__ w_hh,
                            const float* __restrict__ b_ih, const float* __restrict__ b_hh,
                            float* __restrict__ hout) {
    __shared__ float h[128], c[128], xt[32];
    int b = blockIdx.x;
    int t = threadIdx.x;  // 128 threads, one per hidden unit
    h[t] = 0.f; c[t] = 0.f;
    __syncthreads();
    for (int step = 0; step < 30; ++step) {
        int idx = qidx[b * 30 + step];
        if (t < 32) xt[t] = emb[idx * 32 + t];
        __syncthreads();
        float z[4];
#pragma unroll
        for (int g = 0; g < 4; ++g) {
            int row = g * 128 + t;
            float acc = b_ih[row] + b_hh[row];
            const float* wi = w_ih + row * 32;
            for (int k = 0; k < 32; ++k)  acc = fmaf(wi[k], xt[k], acc);
            const float* wh = w_hh + row * 128;
            for (int k = 0; k < 128; ++k) acc = fmaf(wh[k], h[k], acc);
            z[g] = acc;
        }
        float ig = 1.f / (1.f + expf(-z[0]));
        float fg = 1.f / (1.f + expf(-z[1]));
        float gg = tanhf(z[2]);
        float og = 1.f / (1.f + expf(-z[3]));
        float cn = fg * c[t] + ig * gg;
        float hn = og * tanhf(cn);
        __syncthreads();           // all reads of h done
        h[t] = hn; c[t] = cn;
        __syncthreads();
    }
    hout[b * 128 + t] = h[t];
}

// ---------------- generic C = [relu](X @ W^T + bias), WMMA f32 -------------
// X: [Mtiles*16, K] row-major ld=ldX ; W: [Ntiles*16, K] row-major ld=ldW.
// One wave computes one 16x16 output tile via V_WMMA_F32_16X16X4_F32.
// A layout: lane(l16) = row M, half selects K pair {0,1}/{2,3} in 2 VGPRs.
// B layout mirrors it with lane = column N. D: lane = N, vgpr v -> M = half*8+v.
__global__ void gemm_xwT_wmma(const float* __restrict__ X, int ldX,
                              const float* __restrict__ W, int ldW,
                              const float* __restrict__ bias,
                              float* __restrict__ C, int ldC,
                              int Mtiles, int Ntiles, int K, int relu) {
    int wave = (blockIdx.x * blockDim.x + threadIdx.x) >> 5;
    int lane = threadIdx.x & 31;
    int tn = wave % Ntiles;
    int tm = wave / Ntiles;
    if (tm >= Mtiles) return;                 // wave-uniform exit (EXEC all-1 inside)
    int half = lane >> 4;
    int l16  = lane & 15;
    const float* ap = X + (size_t)(tm * 16 + l16) * ldX + half * 2;
    const float* wp = W + (size_t)(tn * 16 + l16) * ldW + half * 2;
    v8f acc = {0.f, 0.f, 0.f, 0.f, 0.f, 0.f, 0.f, 0.f};
    for (int k = 0; k < K; k += 4) {
        v2f a = *(const v2f*)(ap + k);
        v2f b = *(const v2f*)(wp + k);
        acc = __builtin_amdgcn_wmma_f32_16x16x4_f32(false, a, false, b,
                                                    (short)0, acc, false, false);
    }
    int n = tn * 16 + l16;
    float bv = bias ? bias[n] : 0.f;
#pragma unroll
    for (int v = 0; v < 8; ++v) {
        float val = acc[v] + bv;
        if (relu) val = fmaxf(val, 0.f);
        C[(size_t)(tm * 16 + half * 8 + v) * ldC + n] = val;
    }
}

// ------------- pair reduction: xg[b,o] = sum_ij relu(A_i+B_j+Qc) -----------
__global__ void pair_reduce(const float* __restrict__ A, const float* __restrict__ Bm,
                            const float* __restrict__ Qc, float* __restrict__ out) {
    __shared__ float Bs[32 * 256];            // 32 KB tile of B
    int b = blockIdx.x;
    int t = threadIdx.x;                      // 256 threads = channel o
    const float* Ab = A  + (size_t)b * 64 * 256;
    const float* Bb = Bm + (size_t)b * 64 * 256;
    float q = Qc[b * 256 + t];
    float acc = 0.f;
    for (int jt = 0; jt < 2; ++jt) {
        for (int r = 0; r < 32; ++r)
            Bs[r * 256 + t] = Bb[(jt * 32 + r) * 256 + t];
        __syncthreads();
        for (int i = 0; i < 64; ++i) {
            float aq = Ab[i * 256 + t] + q;
#pragma unroll 8
            for (int j = 0; j < 32; ++j)
                acc += fmaxf(aq + Bs[j * 256 + t], 0.f);
        }
        __syncthreads();
    }
    out[b * 256 + t] = acc;
}

// ----------------- f3 linear + log_softmax (wave32 reductions) -------------
__global__ void f3_logsoftmax(const float* __restrict__ xf2, const float* __restrict__ W,
                              const float* __restrict__ bias, float* __restrict__ out) {
    int row = blockIdx.x;
    int t = threadIdx.x;                      // 32 threads = 1 wave
    float logit = -1e30f;
    if (t < 28) {
        float acc = bias[t];
        const float* x = xf2 + row * 256;
        const float* w = W + t * 256;
        for (int k = 0; k < 256; ++k) acc = fmaf(x[k], w[k], acc);
        logit = acc;
    }
    float mx = logit;
#pragma unroll
    for (int off = 16; off > 0; off >>= 1)
        mx = fmaxf(mx, __shfl_xor(mx, off, 32));
    float e = (t < 28) ? expf(logit - mx) : 0.f;
    float s = e;
#pragma unroll
    for (int off = 16; off > 0; off >>= 1)
        s += __shfl_xor(s, off, 32);
    if (t < 28) out[row * 28 + t] = logit - mx - logf(s);
}

// ---------------------------------------------------------------------------
extern "C" void kernel_launch(void* const* d_in, const int* in_sizes, int n_in,
                              void* d_out, int out_size, void* d_ws, size_t ws_size,
                              hipStream_t stream) {
    (void)in_sizes; (void)n_in; (void)out_size; (void)ws_size;
    const float* img  = (const float*)d_in[0];
    const int*   qidx = (const int*)  d_in[1];
    const float* cw1  = (const float*)d_in[2];
    const float* cb1  = (const float*)d_in[3];
    const float* cw   = (const float*)d_in[4];
    const float* cb   = (const float*)d_in[5];
    const float* bng  = (const float*)d_in[6];
    const float* bnb  = (const float*)d_in[7];
    const float* bnm  = (const float*)d_in[8];
    const float* bnv  = (const float*)d_in[9];
    const float* emb  = (const float*)d_in[10];
    const float* w_ih = (const float*)d_in[11];
    const float* w_hh = (const float*)d_in[12];
    const float* b_ih = (const float*)d_in[13];
    const float* b_hh = (const float*)d_in[14];
    const float* g0_w = (const float*)d_in[15];
    const float* g0_b = (const float*)d_in[16];
    const float* f1_w = (const float*)d_in[17];
    const float* f1_b = (const float*)d_in[18];
    const float* f2_w = (const float*)d_in[19];
    const float* f2_b = (const float*)d_in[20];
    const float* f3_w = (const float*)d_in[21];
    const float* f3_b = (const float*)d_in[22];
    float* out = (float*)d_out;
    float* ws  = (float*)d_ws;

    // workspace layout (floats)
    float* c1  = ws;                   // 64*24*64*64 = 6,291,456
    float* c2  = c1  + 6291456;        // 1,572,864
    float* c3  = c2  + 1572864;        //   393,216
    float* c4  = c3  + 393216;         //    98,304
    float* obj = c4  + 98304;          //   114,688
    float* wga = obj + 114688;         //     7,168
    float* wgb = wga + 7168;           //     7,168
    float* qh  = wgb + 7168;           //     8,192
    float* qc  = qh  + 8192;           //    16,384
    float* xg  = qc  + 16384;          //    16,384
    float* x1  = xg  + 16384;          //    16,384
    float* x2  = x1  + 16384;          //    16,384
    // A/B projections alias the c1 region (c1 dead after conv2)
    float* Ao  = ws;                   // 1,048,576
    float* Bo  = ws + 1048576;         // 1,048,576

    // conv stack
    {   int total = 64 * 24 * 64 * 64;
        conv_bn_relu<<<(total + 255) / 256, 256, 0, stream>>>(
            img, cw1, cb1, bng, bnb, bnm, bnv, c1, 3, 24, 128, 128, 64, 64, 64); }
    {   int total = 64 * 24 * 32 * 32;
        conv_bn_relu<<<(total + 255) / 256, 256, 0, stream>>>(
            c1, cw, cb, bng + 24, bnb + 24, bnm + 24, bnv + 24,
            c2, 24, 24, 64, 64, 32, 32, 64); }
    {   int total = 64 * 24 * 16 * 16;
        conv_bn_relu<<<(total + 255) / 256, 256, 0, stream>>>(
            c2, cw + 5184, cb + 24, bng + 48, bnb + 48, bnm + 48, bnv + 48,
            c3, 24, 24, 32, 32, 16, 16, 64); }
    {   int total = 64 * 24 * 8 * 8;
        conv_bn_relu<<<(total + 255) / 256, 256, 0, stream>>>(
            c3, cw + 10368, cb + 48, bng + 72, bnb + 72, bnm + 72, bnv + 72,
            c4, 24, 24, 16, 16, 8, 8, 64); }

    assemble_obj<<<(4096 * 28 + 255) / 256, 256, 0, stream>>>(c4, obj);
    pad_gw<<<(256 * 28 + 255) / 256, 256, 0, stream>>>(g0_w, wga, wgb);
    lstm_kernel<<<64, 128, 0, stream>>>(qidx, emb, w_ih, w_hh, b_ih, b_hh, qh);

    // Qc = qh @ g0_w[:,52:180]^T + g0_b   (K=128, W read in-place, ld=180)
    gemm_xwT_wmma<<<16, 128, 0, stream>>>(qh, 128, g0_w + 52, 180, g0_b,
                                          qc, 256, 4, 16, 128, 0);
    // A,B object projections (K padded to 28): 256x16 tiles each
    gemm_xwT_wmma<<<1024, 128, 0, stream>>>(obj, 28, wga, 28, nullptr,
                                            Ao, 256, 256, 16, 28, 0);
    gemm_xwT_wmma<<<1024, 128, 0, stream>>>(obj, 28, wgb, 28, nullptr,
                                            Bo, 256, 256, 16, 28, 0);

    pair_reduce<<<64, 256, 0, stream>>>(Ao, Bo, qc, xg);

    gemm_xwT_wmma<<<16, 128, 0, stream>>>(xg, 256, f1_w, 256, f1_b,
                                          x1, 256, 4, 16, 256, 1);
    gemm_xwT_wmma<<<16, 128, 0, stream>>>(x1, 256, f2_w, 256, f2_b,
                                          x2, 256, 4, 16, 256, 1);
    f3_logsoftmax<<<64, 32, 0, stream>>>(x2, f3_w, f3_b, out);
}